// LSHLayer_77068893160002
// MI455X (gfx1250) — compile-verified
//
#include <hip/hip_runtime.h>
#include <hip/hip_bf16.h>

typedef __attribute__((ext_vector_type(2))) float v2f;
typedef __attribute__((ext_vector_type(4))) float v4f;
typedef __attribute__((ext_vector_type(8))) float v8f;
typedef __attribute__((ext_vector_type(4))) int   v4i;

#define N_ROWS   2048
#define D_DIM    1024
#define OUT_DIM  65536
#define S_COLS   16384

#define BM 128            // rows per workgroup
#define BN 64             // gathered cols per workgroup
#define KB 32             // K block staged in LDS
#define NKB (D_DIM / KB)  // 32 K-blocks
#define LDSW 36           // padded stride: 144B rows (16B aligned), 36*l mod 64 distinct -> conflict-free

#if defined(__gfx1250__) && __has_builtin(__builtin_amdgcn_global_load_async_to_lds_b128)
#define HAVE_ASYNC 1
#else
#define HAVE_ASYNC 0
#endif

#if HAVE_ASYNC
typedef __attribute__((address_space(1))) v4i* gv4i_p;   // global (AS1) int4*
typedef __attribute__((address_space(3))) v4i* lv4i_p;   // LDS    (AS3) int4*
#endif

__device__ __forceinline__ void copy_b128(const float* g, float* l) {
#if HAVE_ASYNC
    // GLOBAL_LOAD_ASYNC_TO_LDS_B128: per-lane 16B global -> LDS, tracked by ASYNCcnt
    __builtin_amdgcn_global_load_async_to_lds_b128(
        (gv4i_p)(g), (lv4i_p)(l), /*offset=*/0, /*cpol=*/0);
#else
    *(v4f*)l = *(const v4f*)g;   // 16B aligned at stride 36
#endif
}

__device__ __forceinline__ void wait_async() {
#if HAVE_ASYNC
#if __has_builtin(__builtin_amdgcn_s_wait_asynccnt)
    __builtin_amdgcn_s_wait_asynccnt(0);
#else
    asm volatile("s_wait_asynccnt 0x0" ::: "memory");
#endif
#endif
}

__global__ __launch_bounds__(256) void lsh_wmma_f32_kernel(
    const float* __restrict__ x,
    const float* __restrict__ weight,
    const float* __restrict__ bias,
    const int*   __restrict__ sample_ids,
    float*       __restrict__ out)
{
    __shared__ __align__(16) float ldsX[2][BM * LDSW];  // double-buffered x tile
    __shared__ __align__(16) float ldsW[2][BN * LDSW];  // double-buffered gathered W tile
    __shared__ int   lds_sid[BN];
    __shared__ float lds_bias[BN];

    const int tid  = threadIdx.x;
    const int wave = tid >> 5;           // 0..7 (wave32)
    const int lane = tid & 31;
    const int l    = lane & 15;
    const int h    = lane >> 4;

    const int colBase = blockIdx.x * BN; // S tile
    const int rowBase = blockIdx.y * BM; // N tile

    if (tid < BN) {
        const int sid = sample_ids[colBase + tid];
        lds_sid[tid]  = sid;
        lds_bias[tid] = bias[sid];
    }
    __syncthreads();

    // Stage one K-block (BM x KB of x, gathered BN x KB of W) into LDS buffer `buf`.
    auto stage = [&](int kb, int buf) {
        const int k0 = kb * KB;
        #pragma unroll
        for (int i = 0; i < 4; ++i) {                 // 128 rows x 32 K = 1024 f4
            const int idx = tid + i * 256;
            const int r   = idx >> 3;                 // row 0..127
            const int kc  = (idx & 7) << 2;           // 0,4,...,28
            copy_b128(x + (size_t)(rowBase + r) * D_DIM + (k0 + kc),
                      &ldsX[buf][r * LDSW + kc]);
        }
        #pragma unroll
        for (int i = 0; i < 2; ++i) {                 // 64 cols x 32 K = 512 f4
            const int idx = tid + i * 256;
            const int c   = idx >> 3;                 // col 0..63
            const int kc  = (idx & 7) << 2;
            copy_b128(weight + (size_t)lds_sid[c] * D_DIM + (k0 + kc),
                      &ldsW[buf][c * LDSW + kc]);
        }
    };

    v8f acc[4];
    #pragma unroll
    for (int t = 0; t < 4; ++t) acc[t] = (v8f){};

    stage(0, 0);                                      // prologue fill

    for (int kb = 0; kb < NKB; ++kb) {
        const int cur = kb & 1;
        wait_async();                                 // my async copies for `cur` done
        __syncthreads();                              // everyone's copies done / prev reads done

        if (kb + 1 < NKB) stage(kb + 1, cur ^ 1);     // overlap next fill with compute

        #pragma unroll
        for (int ks = 0; ks < KB; ks += 4) {
            const int kf = ks + 2 * h;                // K = 2h + component
            const float* ax = &ldsX[cur][(wave * 16 + l) * LDSW + kf];
            v2f a; a.x = ax[0]; a.y = ax[1];          // A: M=l, K=kf,kf+1
            #pragma unroll
            for (int t = 0; t < 4; ++t) {
                const float* bx = &ldsW[cur][(t * 16 + l) * LDSW + kf];
                v2f b; b.x = bx[0]; b.y = bx[1];      // B: N=l, K=kf,kf+1
                acc[t] = __builtin_amdgcn_wmma_f32_16x16x4_f32(
                    false, a, false, b, (short)0, acc[t], false, false);
            }
        }
    }

    // Epilogue: C/D layout VGPR r, half h -> M = r + 8h ; N = lane%16
    #pragma unroll
    for (int t = 0; t < 4; ++t) {
        const float bb   = lds_bias[t * 16 + l];
        const size_t col = (size_t)colBase + t * 16 + l;
        #pragma unroll
        for (int r = 0; r < 8; ++r) {
            const size_t row = (size_t)rowBase + wave * 16 + 8 * h + r;
            out[row * S_COLS + col] = acc[t][r] + bb;
        }
    }
}

extern "C" void kernel_launch(void* const* d_in, const int* in_sizes, int n_in,
                              void* d_out, int out_size, void* d_ws, size_t ws_size,
                              hipStream_t stream) {
    const float* x          = (const float*)d_in[0];
    const float* weight     = (const float*)d_in[1];
    const float* bias       = (const float*)d_in[2];
    const int*   sample_ids = (const int*)  d_in[3];
    float*       out        = (float*)d_out;

    dim3 grid(S_COLS / BN, N_ROWS / BM);   // 256 x 16 workgroups
    dim3 block(256);                        // 8 waves (wave32)
    lsh_wmma_f32_kernel<<<grid, block, 0, stream>>>(x, weight, bias, sample_ids, out);
    (void)in_sizes; (void)n_in; (void)out_size; (void)d_ws; (void)ws_size;
}